// AutoFeedBack_89885075570792
// MI455X (gfx1250) — compile-verified
//
#include <hip/hip_runtime.h>
#include <math.h>

// ---------------- problem constants ----------------
#define UNITS     1024
#define N3        3072              // 3 * UNITS
#define OUT_STEPS 400
#define SEQ       (4096 + OUT_STEPS)   // 4496
#define NF        4
#define WARM      4096
#define NSTEPS    (WARM + OUT_STEPS - 1)  // 4495 sequential cell steps

// ---------------- phase-2 persistent-kernel geometry ----------------
#define G    64        // workgroups; each owns 16 rows of h (48 gate columns)
#define TPB  192       // 6 wave32 waves: thread = (c:16) x (gate:3) x (p:4 k-parts)
#define KPART 256      // k elements per partial (4 * 256 = 1024)

// ---------------- workspace layout (bytes) ----------------
#define MX_BYTES   ((size_t)SEQ * N3 * sizeof(float))   // 55.2 MB: hoisted input GEMM
#define H_OFF      (MX_BYTES)
#define H_BYTES    (2u * UNITS * sizeof(float))         // double-buffered hidden state
#define PART_OFF   (H_OFF + H_BYTES)
#define PART_BYTES (2u * G * sizeof(float))             // double-buffered dense partials
#define BAR_OFF    (PART_OFF + PART_BYTES)              // 2 x u32: count, generation

typedef __attribute__((ext_vector_type(2))) float v2f;
typedef __attribute__((ext_vector_type(8))) float v8f;

#if defined(__has_builtin)
#if __has_builtin(__builtin_amdgcn_wmma_f32_16x16x4_f32)
#define HAVE_WMMA_F32X4 1
#endif
#if __has_builtin(__builtin_amdgcn_global_load_async_to_lds_b32)
#define HAVE_ASYNC_LDS 1
#endif
#endif

typedef __attribute__((address_space(1))) int as1_int;   // global (prints as __device__)
typedef __attribute__((address_space(3))) int as3_int;   // LDS

// ============ init: zero h0 and the global barrier state ============
__global__ void init_ws(float* __restrict__ h, unsigned* __restrict__ bar) {
    for (int i = threadIdx.x; i < 2 * UNITS; i += blockDim.x) h[i] = 0.0f;
    if (threadIdx.x < 2) bar[threadIdx.x] = 0u;
}

// ============ phase 1: MX = X @ K + bi  via V_WMMA_F32_16X16X4_F32 ============
// M = 4496 = 281 tiles of 16, N = 3072 = 192 tiles of 16, K = NF = 4 exactly.
// One 16x16 output tile per wave; 8 waves (256 threads) per block; no tails.
__global__ __launch_bounds__(256) void mx_gemm(
    const float* __restrict__ X,     // [SEQ, 4]
    const float* __restrict__ K,     // [4, 3072]
    const float* __restrict__ bias,  // [2, 3072]; row 0 = bi
    float* __restrict__ MX)          // [SEQ, 3072]
{
    const int wid  = threadIdx.x >> 5;
    const int lane = threadIdx.x & 31;
    const int tile = blockIdx.x * 8 + wid;         // exact: 6744*8 == 281*192
    const int tm = tile / (N3 / 16);
    const int tn = tile % (N3 / 16);
    const int t0 = tm * 16;
    const int n0 = tn * 16;

    const int m    = lane & 15;     // A row / B,C,D column-within-tile
    const int half = lane >> 4;     // lane group select
    const int n    = n0 + m;

#if HAVE_WMMA_F32X4
    // A 16x4 f32: VGPR0 = K-col {0|2}, VGPR1 = K-col {1|3} per lane half
    v2f a;
    a.x = X[(t0 + m) * NF + 2 * half];
    a.y = X[(t0 + m) * NF + 2 * half + 1];
    // B 4x16 f32: VGPR0 = K-row {0|2}, VGPR1 = K-row {1|3} per lane half
    v2f b;
    b.x = K[(2 * half) * N3 + n];
    b.y = K[(2 * half + 1) * N3 + n];
    // C = input bias broadcast down the column
    const float bi = bias[n];
    v8f c;
    #pragma unroll
    for (int i = 0; i < 8; ++i) c[i] = bi;

    v8f d = __builtin_amdgcn_wmma_f32_16x16x4_f32(
        /*neg_a=*/false, a, /*neg_b=*/false, b,
        /*c_mod=*/(short)0, c, /*reuse_a=*/false, /*reuse_b=*/false);

    // D 16x16 f32: VGPR i holds rows i (lanes 0-15) and i+8 (lanes 16-31)
    #pragma unroll
    for (int i = 0; i < 8; ++i)
        MX[(t0 + i + 8 * half) * N3 + n] = d[i];
#else
    // scalar fallback (signals wmma=0 in histogram if builtin is absent)
    for (int i = 0; i < 8; ++i) {
        const int t = t0 + i + 8 * half;
        float acc = bias[n];
        #pragma unroll
        for (int f = 0; f < NF; ++f) acc = fmaf(X[t * NF + f], K[f * N3 + n], acc);
        MX[t * N3 + n] = acc;
    }
#endif
}

// ============ phase 2: persistent sequential GRU ============
// WG g owns h rows j in [16g, 16g+16) -> gate columns {j, 1024+j, 2048+j}.
// Thread (c,gate,p): 256-long k-partial of dot(h, R[:, gate*1024+16g+c]).
// One device-scope barrier per step; dense head reduced in fixed order.
__global__ __launch_bounds__(TPB) void gru_seq(
    const float* __restrict__ X,    // [SEQ, 4]
    const float* __restrict__ K,    // [4, 3072]  (row 3 = feedback-feature column)
    const float* __restrict__ R,    // [1024, 3072]
    const float* __restrict__ bias, // [2, 3072]  (row 1 = br)
    const float* __restrict__ dw,   // [1024]
    const float* __restrict__ db,   // [1]
    const float* __restrict__ MX,   // [SEQ, 3072] from phase 1
    float* __restrict__ hbuf,       // [2, 1024]
    float* __restrict__ part,       // [2, G]
    unsigned* __restrict__ bar,     // {count, generation}
    float* __restrict__ out)        // [OUT_STEPS]
{
    __shared__ __attribute__((aligned(16))) float h_sh[UNITS];
    __shared__ float red[TPB];
    __shared__ float mh_sh[48];
    __shared__ float k3_sh[48];
    __shared__ float br_sh[48];
    __shared__ float dw_sh[16];
    __shared__ float pd_sh[16];

    const int tid = threadIdx.x;
    const int wg  = blockIdx.x;
    const int c    = tid & 15;
    const int gate = (tid >> 4) % 3;
    const int p    = tid / 48;
    const int n    = gate * UNITS + wg * 16 + c;   // this thread's gate column

    if (tid < 48) {
        const int nn = (tid >> 4) * UNITS + wg * 16 + (tid & 15);
        k3_sh[tid] = K[3 * N3 + nn];   // feedback-feature row of the input kernel
        br_sh[tid] = bias[N3 + nn];    // recurrent bias
    }
    if (tid < 16) dw_sh[tid] = dw[wg * 16 + tid];
    const float dbv = db[0];

    unsigned mygen = 0u;
    float pred = 0.0f;

    const float* __restrict__ Rcol = R + n;

    for (int step = 0; step < NSTEPS; ++step) {
        const int  t  = (step < WARM) ? step : (WARM + 1 + (step - WARM));
        const bool ar = (step >= WARM);
        const float* __restrict__ hg = hbuf + (step & 1) * UNITS;
        float* __restrict__       hn = hbuf + ((step + 1) & 1) * UNITS;

        // warm this step's MX gate segments while the matvec runs
        if (tid == 0) {
            __builtin_prefetch(&MX[t * N3 + wg * 16], 0, 0);
            __builtin_prefetch(&MX[t * N3 + UNITS + wg * 16], 0, 0);
            __builtin_prefetch(&MX[t * N3 + 2 * UNITS + wg * 16], 0, 0);
        }

        // broadcast h into LDS (async global->LDS path when available)
#if HAVE_ASYNC_LDS
        for (int i = tid; i < UNITS; i += TPB) {
            __builtin_amdgcn_global_load_async_to_lds_b32(
                (as1_int*)(hg + i), (as3_int*)&h_sh[i], 0, 0);
        }
  #if defined(__has_builtin) && __has_builtin(__builtin_amdgcn_s_wait_asynccnt)
        __builtin_amdgcn_s_wait_asynccnt(0);
  #else
        asm volatile("s_wait_asynccnt 0x0" ::: "memory");
  #endif
        __syncthreads();
#else
        for (int i = tid; i < UNITS; i += TPB) h_sh[i] = hg[i];
        __syncthreads();
#endif

        // k-partial of the recurrent matvec: 4 independent accumulator chains,
        // 16-lane 64B-coalesced column-group loads from the L2-resident R.
        float a0 = 0.0f, a1 = 0.0f, a2 = 0.0f, a3 = 0.0f;
        const float* __restrict__ rp = Rcol + p * KPART * N3;
        const float* __restrict__ hp = h_sh + p * KPART;
        #pragma unroll 4
        for (int kk = 0; kk < KPART; kk += 4) {
            a0 = fmaf(hp[kk + 0], rp[(kk + 0) * N3], a0);
            a1 = fmaf(hp[kk + 1], rp[(kk + 1) * N3], a1);
            a2 = fmaf(hp[kk + 2], rp[(kk + 2) * N3], a2);
            a3 = fmaf(hp[kk + 3], rp[(kk + 3) * N3], a3);
        }
        red[tid] = (a0 + a1) + (a2 + a3);
        __syncthreads();

        if (tid < 48)
            mh_sh[tid] = red[tid] + red[tid + 48] + red[tid + 96] + red[tid + 144] + br_sh[tid];
        __syncthreads();

        if (tid < 16) {
            const int j = wg * 16 + tid;
            float mxz = MX[t * N3 + j];
            float mxr = MX[t * N3 + UNITS + j];
            float mxh = MX[t * N3 + 2 * UNITS + j];
            if (ar) {  // rank-1 fix: swap precomputed feature 3 for the fed-back pred
                const float dx = pred - X[t * NF + 3];
                mxz = fmaf(dx, k3_sh[tid],      mxz);
                mxr = fmaf(dx, k3_sh[16 + tid], mxr);
                mxh = fmaf(dx, k3_sh[32 + tid], mxh);
            }
            const float z  = 1.0f / (1.0f + __expf(-(mxz + mh_sh[tid])));
            const float r  = 1.0f / (1.0f + __expf(-(mxr + mh_sh[16 + tid])));
            const float hh = tanhf(fmaf(r, mh_sh[32 + tid], mxh));
            const float hv = z * h_sh[j] + (1.0f - z) * hh;
            hn[j] = hv;
            pd_sh[tid] = hv * dw_sh[tid];
        }
        __syncthreads();

        if (step >= WARM - 1 && tid == 0) {
            float s = 0.0f;
            #pragma unroll
            for (int i = 0; i < 16; ++i) s += pd_sh[i];
            part[(step & 1) * G + wg] = s;
        }

        // ---- device-scope sense-reversing barrier (one per step) ----
        __threadfence();
        __syncthreads();
        if (tid == 0) {
            const unsigned target = mygen + 1u;
            if (__hip_atomic_fetch_add(&bar[0], 1u, __ATOMIC_ACQ_REL,
                                       __HIP_MEMORY_SCOPE_AGENT) == G - 1u) {
                __hip_atomic_store(&bar[0], 0u, __ATOMIC_RELAXED, __HIP_MEMORY_SCOPE_AGENT);
                __hip_atomic_fetch_add(&bar[1], 1u, __ATOMIC_RELEASE, __HIP_MEMORY_SCOPE_AGENT);
            } else {
                while (__hip_atomic_load(&bar[1], __ATOMIC_ACQUIRE,
                                         __HIP_MEMORY_SCOPE_AGENT) < target)
                    __builtin_amdgcn_s_sleep(1);
            }
            mygen = target;
        }
        __syncthreads();

        if (step >= WARM - 1) {
            // every WG sums the 64 partials in the same fixed order -> deterministic
            const float* __restrict__ pp = part + (step & 1) * G;
            float s = dbv;
            for (int i = 0; i < G; ++i) s += pp[i];
            pred = 1.0f / (1.0f + __expf(-s));
            if (wg == 0 && tid == 0) out[step - (WARM - 1)] = pred;
        }
    }
}

// ============ launcher ============
extern "C" void kernel_launch(void* const* d_in, const int* in_sizes, int n_in,
                              void* d_out, int out_size, void* d_ws, size_t ws_size,
                              hipStream_t stream) {
    (void)in_sizes; (void)n_in; (void)out_size; (void)ws_size;
    const float* X    = (const float*)d_in[0];  // inputs  [1, SEQ, 4]
    const float* K    = (const float*)d_in[1];  // kernel  [4, 3072]
    const float* R    = (const float*)d_in[2];  // recurrent_kernel [1024, 3072]
    const float* bias = (const float*)d_in[3];  // [2, 3072]
    const float* dw   = (const float*)d_in[4];  // [1024, 1]
    const float* db   = (const float*)d_in[5];  // [1]
    float* out = (float*)d_out;

    char* ws = (char*)d_ws;
    float*    MX   = (float*)(ws);
    float*    hbuf = (float*)(ws + H_OFF);
    float*    part = (float*)(ws + PART_OFF);
    unsigned* bar  = (unsigned*)(ws + BAR_OFF);

    init_ws<<<1, 256, 0, stream>>>(hbuf, bar);

    const int tiles  = (SEQ / 16) * (N3 / 16);   // 281 * 192 = 53952, exact
    const int blocks = tiles / 8;                // 6744, exact
    mx_gemm<<<blocks, 256, 0, stream>>>(X, K, bias, MX);

    gru_seq<<<G, TPB, 0, stream>>>(X, K, R, bias, dw, db, MX, hbuf, part, bar, out);
}